// GIN_37606733644137
// MI455X (gfx1250) — compile-verified
//
#include <hip/hip_runtime.h>

// GIN (2x GINConv 'max', eps=0) for MI455X / gfx1250.
// Edge phases: order-preserving float->u32 encode + global_atomic_max_u32 (L2-resident).
// Node phases: fp32 WMMA (v_wmma_f32_16x16x4_f32) GEMM, bias (+relu for layer 1).

typedef __attribute__((ext_vector_type(2))) float v2f;
typedef __attribute__((ext_vector_type(8))) float v8f;

#define IN_FEATS 64
#define H_FEATS 128
#define NUM_CLASSES 32

// Order-preserving map: float -> unsigned, so unsigned-max == float-max.
__device__ __forceinline__ unsigned enc_f32(float x) {
    unsigned b = __float_as_uint(x);
    return (b & 0x80000000u) ? ~b : (b | 0x80000000u);
}
// Sentinel 0u (strictly below every finite encoding) == "empty segment" -> 0.0f,
// matching reference's where(isfinite(agg), agg, 0).
__device__ __forceinline__ float dec_f32(unsigned u) {
    if (u == 0u) return 0.0f;
    unsigned b = (u & 0x80000000u) ? (u & 0x7FFFFFFFu) : ~u;
    return __uint_as_float(b);
}

// One thread per (edge, float4-chunk). Gathers are b128 loads from an
// L2-resident node table; scatter is 4x global_atomic_max_u32 (no return).
__global__ void gin_scatter_max(const float* __restrict__ x,
                                const int* __restrict__ src,
                                const int* __restrict__ dst,
                                const float* __restrict__ ew,
                                unsigned* __restrict__ agg,
                                int nEdges, int feats, int lgGroups) {
    int gid = blockIdx.x * blockDim.x + threadIdx.x;
    int groups = 1 << lgGroups;                 // feats / 4
    if (gid >= nEdges * groups) return;         // <= 38.4M, fits in int
    int e  = gid >> lgGroups;
    int f4 = gid & (groups - 1);
    int s = src[e];
    int d = dst[e];
    float w = ew[e];
    float4 v = reinterpret_cast<const float4*>(x + (size_t)s * feats)[f4];
    unsigned* ap = agg + (size_t)d * feats + f4 * 4;
    atomicMax(ap + 0, enc_f32(v.x * w));
    atomicMax(ap + 1, enc_f32(v.y * w));
    atomicMax(ap + 2, enc_f32(v.z * w));
    atomicMax(ap + 3, enc_f32(v.w * w));
}

// Layer 1: h1[16 rows x 128 cols per block] = relu((x+agg) @ W1^T + b1).
// 8 waves; wave w owns output columns [16w, 16w+16). K=64 -> 16 WMMAs/wave.
__global__ void __launch_bounds__(256)
gin_layer1(const float* __restrict__ x, const unsigned* __restrict__ agg,
           const float* __restrict__ W1, const float* __restrict__ b1,
           float* __restrict__ h1) {
    __shared__ float tile[16 * IN_FEATS];           // combined x + decode(agg)
    const int row0 = blockIdx.x * 16;               // N = 100000 = 6250*16: no tail
    const int tid  = threadIdx.x;
    for (int i = tid; i < 16 * IN_FEATS; i += 256) {
        int g = row0 * IN_FEATS + i;                // tile is row-linear
        tile[i] = x[g] + dec_f32(agg[g]);
    }
    __syncthreads();

    const int wave  = tid >> 5;
    const int lane  = tid & 31;
    const int laneM = lane & 15;
    const int kHi   = (lane >> 4) << 1;             // lanes 0-15: K+{0,1}; 16-31: K+{2,3}
    const int h0    = wave * 16;

    v8f acc = {};
#pragma unroll
    for (int k0 = 0; k0 < IN_FEATS; k0 += 4) {
        v2f a, b;
        a[0] = tile[laneM * IN_FEATS + k0 + kHi];
        a[1] = tile[laneM * IN_FEATS + k0 + kHi + 1];
        // B[k][n] = W1[h0+n][k]  (out = in @ W1^T)
        b[0] = W1[(h0 + laneM) * IN_FEATS + k0 + kHi];
        b[1] = W1[(h0 + laneM) * IN_FEATS + k0 + kHi + 1];
        acc = __builtin_amdgcn_wmma_f32_16x16x4_f32(
            false, a, false, b, (short)0, acc, false, false);
    }

    // C/D layout: lanes 0-15 -> rows 0..7, lanes 16-31 -> rows 8..15; col = laneM.
    const int col   = h0 + laneM;
    const float bia = b1[col];
    const int rbase = row0 + ((lane >> 4) << 3);
#pragma unroll
    for (int i = 0; i < 8; ++i) {
        float v = acc[i] + bia;
        h1[(size_t)(rbase + i) * H_FEATS + col] = v > 0.0f ? v : 0.0f;
    }
}

// Layer 2: out[16 rows x 32 cols per block] = (h1+agg2) @ W2^T + b2 (no relu).
// 2 waves; K=128 -> 32 WMMAs/wave.
__global__ void __launch_bounds__(64)
gin_layer2(const float* __restrict__ h1, const unsigned* __restrict__ agg,
           const float* __restrict__ W2, const float* __restrict__ b2,
           float* __restrict__ out) {
    __shared__ float tile[16 * H_FEATS];
    const int row0 = blockIdx.x * 16;
    const int tid  = threadIdx.x;
    for (int i = tid; i < 16 * H_FEATS; i += 64) {
        int g = row0 * H_FEATS + i;
        tile[i] = h1[g] + dec_f32(agg[g]);
    }
    __syncthreads();

    const int wave  = tid >> 5;
    const int lane  = tid & 31;
    const int laneM = lane & 15;
    const int kHi   = (lane >> 4) << 1;
    const int h0    = wave * 16;

    v8f acc = {};
#pragma unroll
    for (int k0 = 0; k0 < H_FEATS; k0 += 4) {
        v2f a, b;
        a[0] = tile[laneM * H_FEATS + k0 + kHi];
        a[1] = tile[laneM * H_FEATS + k0 + kHi + 1];
        b[0] = W2[(h0 + laneM) * H_FEATS + k0 + kHi];
        b[1] = W2[(h0 + laneM) * H_FEATS + k0 + kHi + 1];
        acc = __builtin_amdgcn_wmma_f32_16x16x4_f32(
            false, a, false, b, (short)0, acc, false, false);
    }

    const int col   = h0 + laneM;
    const float bia = b2[col];
    const int rbase = row0 + ((lane >> 4) << 3);
#pragma unroll
    for (int i = 0; i < 8; ++i) {
        out[(size_t)(rbase + i) * NUM_CLASSES + col] = acc[i] + bia;
    }
}

extern "C" void kernel_launch(void* const* d_in, const int* in_sizes, int n_in,
                              void* d_out, int out_size, void* d_ws, size_t ws_size,
                              hipStream_t stream) {
    const float* x   = (const float*)d_in[0];
    const int*   src = (const int*)d_in[1];
    const int*   dst = (const int*)d_in[2];
    const float* ew  = (const float*)d_in[3];
    const float* W1  = (const float*)d_in[4];
    const float* b1  = (const float*)d_in[5];
    const float* W2  = (const float*)d_in[6];
    const float* b2  = (const float*)d_in[7];
    float* out = (float*)d_out;

    const int N = in_sizes[0] / IN_FEATS;   // 100000 (divisible by 16)
    const int E = in_sizes[1];              // 1200000

    // Workspace layout: [h1 : N*128 f32][agg : N*128 u32]; agg1 aliases agg2
    // (agg1 is dead before the layer-2 memset, stream order makes this safe).
    float*    h1   = (float*)d_ws;
    unsigned* agg2 = (unsigned*)((char*)d_ws + (size_t)N * H_FEATS * sizeof(float));
    unsigned* agg1 = agg2;

    // ---- Layer 1 ----
    hipMemsetAsync(agg1, 0, (size_t)N * IN_FEATS * sizeof(unsigned), stream);
    {
        long long work = (long long)E << 4;          // 16 float4 chunks / edge
        int blocks = (int)((work + 255) / 256);
        gin_scatter_max<<<blocks, 256, 0, stream>>>(x, src, dst, ew, agg1,
                                                    E, IN_FEATS, 4);
    }
    gin_layer1<<<N / 16, 256, 0, stream>>>(x, agg1, W1, b1, h1);

    // ---- Layer 2 ----
    hipMemsetAsync(agg2, 0, (size_t)N * H_FEATS * sizeof(unsigned), stream);
    {
        long long work = (long long)E << 5;          // 32 float4 chunks / edge
        int blocks = (int)((work + 255) / 256);
        gin_scatter_max<<<blocks, 256, 0, stream>>>(h1, src, dst, ew, agg2,
                                                    E, H_FEATS, 5);
    }
    gin_layer2<<<N / 16, 64, 0, stream>>>(h1, agg2, W2, b2, out);
}